// GraphSAGE_68092411510980
// MI455X (gfx1250) — compile-verified
//
#include <hip/hip_runtime.h>
#include <hip/hip_bf16.h>

typedef float v2f __attribute__((ext_vector_type(2)));
typedef float v8f __attribute__((ext_vector_type(8)));

#define FDIM 128   // feature dim of every A source (x, h1, agg)
#define KTOT 384   // 3 * 128 concatenated K
#define KC   64    // K-chunk staged in LDS

// ---------------------------------------------------------------- utilities
__global__ void fill_zero(float* p, size_t n) {
  size_t i = (size_t)blockIdx.x * blockDim.x + threadIdx.x;
  size_t stride = (size_t)gridDim.x * blockDim.x;
  for (; i < n; i += stride) p[i] = 0.0f;
}

// Build stacked weight matrices:
//  Bcat1[384,128] = [Wself1[0]+Wself1[1]; Wneigh1[0]; Wneigh1[1]], bias1=b1[0]+b1[1]
//  Bcat2[384, 64] = [Wself2[0]+Wself2[1]; Wneigh2[0]; Wneigh2[1]], bias2=b2[0]+b2[1]
__global__ void build_weights(const float* __restrict__ Wn1, const float* __restrict__ Ws1,
                              const float* __restrict__ b1,
                              const float* __restrict__ Wn2, const float* __restrict__ Ws2,
                              const float* __restrict__ b2,
                              float* __restrict__ Bcat1, float* __restrict__ bias1,
                              float* __restrict__ Bcat2, float* __restrict__ bias2) {
  int i = blockIdx.x * blockDim.x + threadIdx.x;
  const int S1 = 384 * 128, S2 = S1 + 128, S3 = S2 + 384 * 64, S4 = S3 + 64;
  if (i < S1) {
    int k = i >> 7, n = i & 127;
    float v;
    if (k < 128)      v = Ws1[k * 128 + n] + Ws1[16384 + k * 128 + n];
    else if (k < 256) v = Wn1[(k - 128) * 128 + n];
    else              v = Wn1[16384 + (k - 256) * 128 + n];
    Bcat1[i] = v;
  } else if (i < S2) {
    int n = i - S1;
    bias1[n] = b1[n] + b1[128 + n];
  } else if (i < S3) {
    int j = i - S2;
    int k = j >> 6, n = j & 63;
    float v;
    if (k < 128)      v = Ws2[k * 64 + n] + Ws2[8192 + k * 64 + n];
    else if (k < 256) v = Wn2[(k - 128) * 64 + n];
    else              v = Wn2[8192 + (k - 256) * 64 + n];
    Bcat2[j] = v;
  } else if (i < S4) {
    int n = i - S3;
    bias2[n] = b2[n] + b2[64 + n];
  }
}

// ---------------------------------------------------------------- degrees
__global__ void degree_count(const int* __restrict__ edges, float* deg, int E, int N) {
  int e = blockIdx.x * blockDim.x + threadIdx.x;
  int r = blockIdx.y;
  if (e >= E) return;
  int d = edges[(size_t)r * 2 * E + E + e];
  atomicAdd(deg + (size_t)r * N + d, 1.0f);
}

__global__ void recip_deg(float* deg, int n) {
  int i = blockIdx.x * blockDim.x + threadIdx.x;
  if (i < n) deg[i] = 1.0f / fmaxf(deg[i], 1.0f);   // matches s / max(deg,1)
}

// ---------------------------------------------------------------- scatter (mean aggregate)
// one wave32 per edge, 4 floats per lane (128-wide rows), f32 global atomics
__global__ void scatter_feat(const int* __restrict__ src, const int* __restrict__ dst,
                             const float* __restrict__ feat, float* agg, int E) {
  int gid = blockIdx.x * blockDim.x + threadIdx.x;
  int e = gid >> 5, lane = gid & 31;
  if (e >= E) return;
  int s = src[e], d = dst[e];
  const float4 v = *(const float4*)(feat + (size_t)s * FDIM + lane * 4);
  float* p = agg + (size_t)d * FDIM + lane * 4;
  atomicAdd(p + 0, v.x);
  atomicAdd(p + 1, v.y);
  atomicAdd(p + 2, v.z);
  atomicAdd(p + 3, v.w);
}

__global__ void scale_rows(float* agg, const float* __restrict__ dinv, size_t n) {
  size_t i = (size_t)blockIdx.x * blockDim.x + threadIdx.x;
  if (i < n) agg[i] *= dinv[i >> 7];
}

// ---------------------------------------------------------------- fused WMMA GEMM
// out[N, NOUT] = act( [A0 | G0 | G1] (K=384) @ Bmat[384,NOUT] + bias )
// 256 threads = 8 waves; wave handles 16 rows x NOUT cols; B staged in LDS.
template <int NOUT, bool RELU>
__global__ __launch_bounds__(256) void sage_gemm(
    const float* __restrict__ A0, const float* __restrict__ G0, const float* __restrict__ G1,
    const float* __restrict__ Bmat, const float* __restrict__ bias,
    float* __restrict__ out, int Nrows) {
  constexpr int NT = NOUT / 16;
  __shared__ float Bs[KC * NOUT];

  const int tid = threadIdx.x;
  const int lane = tid & 31;
  const int wave = tid >> 5;
  const int rowBase = blockIdx.x * 128 + wave * 16;
  const int m = lane & 15;
  const int khalf = (lane >> 4) << 1;   // lanes 16-31 carry K+2, K+3

  v8f czero = {};
  v8f c[NT];
#pragma unroll
  for (int t = 0; t < NT; ++t) c[t] = czero;

  // clamp (not branch) so EXEC stays all-ones for WMMA; bogus rows masked at store
  int rowA = rowBase + m;
  if (rowA > Nrows - 1) rowA = Nrows - 1;

  for (int kc = 0; kc < KTOT; kc += KC) {
    for (int i = tid; i < KC * NOUT; i += 256)
      Bs[i] = Bmat[kc * NOUT + i];
    __syncthreads();

    // each K-chunk lies entirely in one A source (128 % 64 == 0)
    const float* Asrc = (kc < 128) ? A0 : ((kc < 256) ? G0 : G1);
    const int kbase = kc & 127;

    for (int k4 = 0; k4 < KC; k4 += 4) {
      // A frag 16x4: lanes0-15 -> {K,K+1}, lanes16-31 -> {K+2,K+3}
      v2f a = *(const v2f*)(Asrc + (size_t)rowA * FDIM + kbase + k4 + khalf);
      const int kb = k4 + khalf;
#pragma unroll
      for (int t = 0; t < NT; ++t) {
        v2f b;
        b.x = Bs[kb * NOUT + t * 16 + m];
        b.y = Bs[(kb + 1) * NOUT + t * 16 + m];
        c[t] = __builtin_amdgcn_wmma_f32_16x16x4_f32(
            false, a, false, b, (short)0, c[t], false, false);
      }
    }
    __syncthreads();
  }

  // C/D layout: VGPR v -> M=v (lanes0-15), M=v+8 (lanes16-31); N = lane&15 per tile
  const int rhalf = (lane >> 4) << 3;
#pragma unroll
  for (int t = 0; t < NT; ++t) {
    int col = t * 16 + m;
    float bv = bias[col];
#pragma unroll
    for (int v = 0; v < 8; ++v) {
      int row = rowBase + v + rhalf;
      if (row < Nrows) {
        float val = c[t][v] + bv;
        if (RELU) val = fmaxf(val, 0.0f);
        out[(size_t)row * NOUT + col] = val;
      }
    }
  }
}

// ---------------------------------------------------------------- edge scoring (O = 64)
__global__ void edge_score(const int* __restrict__ src, const int* __restrict__ dst,
                           const float* __restrict__ h, float* __restrict__ out, int E) {
  int gid = blockIdx.x * blockDim.x + threadIdx.x;
  int e = gid >> 5, lane = gid & 31;
  if (e >= E) return;
  int s = src[e], d = dst[e];
  v2f a = *(const v2f*)(h + (size_t)s * 64 + lane * 2);
  v2f b = *(const v2f*)(h + (size_t)d * 64 + lane * 2);
  float p = a.x * b.x + a.y * b.y;
#pragma unroll
  for (int off = 16; off > 0; off >>= 1)
    p += __shfl_xor(p, off, 32);
  if (lane == 0) out[e] = p;
}

// ---------------------------------------------------------------- launch
extern "C" void kernel_launch(void* const* d_in, const int* in_sizes, int n_in,
                              void* d_out, int out_size, void* d_ws, size_t ws_size,
                              hipStream_t stream) {
  const float* x   = (const float*)d_in[0];
  const int*   edges = (const int*)d_in[1];
  const int*   negE  = (const int*)d_in[2];
  const float* Wn1 = (const float*)d_in[3];
  const float* Ws1 = (const float*)d_in[4];
  const float* b1  = (const float*)d_in[5];
  const float* Wn2 = (const float*)d_in[6];
  const float* Ws2 = (const float*)d_in[7];
  const float* b2  = (const float*)d_in[8];
  float* out = (float*)d_out;

  const int N = in_sizes[0] / 128;
  const int E = in_sizes[1] / 4;   // [R=2, 2, E]

  // workspace layout (~90 MB of f32)
  float* ws    = (float*)d_ws;
  float* dinv  = ws;                         // 2*N  (deg counts -> 1/max(deg,1))
  float* agg0  = dinv + 2 * (size_t)N;       // N*128
  float* agg1  = agg0 + (size_t)N * 128;     // N*128
  float* h1    = agg1 + (size_t)N * 128;     // N*128
  float* h2    = h1 + (size_t)N * 128;       // N*64
  float* Bcat1 = h2 + (size_t)N * 64;        // 384*128
  float* bias1 = Bcat1 + 384 * 128;          // 128
  float* Bcat2 = bias1 + 128;                // 384*64
  float* bias2 = Bcat2 + 384 * 64;           // 64

  const int T = 256;
  const size_t nf = (size_t)N * 128;

  // init + weights
  fill_zero<<<1024, T, 0, stream>>>(dinv, 2 * (size_t)N + 2 * nf);
  const int wtot = 384 * 128 + 128 + 384 * 64 + 64;
  build_weights<<<(wtot + T - 1) / T, T, 0, stream>>>(Wn1, Ws1, b1, Wn2, Ws2, b2,
                                                      Bcat1, bias1, Bcat2, bias2);

  // degrees (shared by both layers)
  degree_count<<<dim3((E + T - 1) / T, 2), T, 0, stream>>>(edges, dinv, E, N);
  recip_deg<<<(2 * N + T - 1) / T, T, 0, stream>>>(dinv, 2 * N);

  const int sblocks = (int)(((size_t)E * 32 + T - 1) / T);
  const int fblocks = (int)((nf + T - 1) / T);
  const int gblocks = (N + 127) / 128;

  // ---- layer 1
  scatter_feat<<<sblocks, T, 0, stream>>>(edges,         edges + E,     x, agg0, E);
  scatter_feat<<<sblocks, T, 0, stream>>>(edges + 2 * E, edges + 3 * E, x, agg1, E);
  scale_rows<<<fblocks, T, 0, stream>>>(agg0, dinv,     nf);
  scale_rows<<<fblocks, T, 0, stream>>>(agg1, dinv + N, nf);
  sage_gemm<128, true><<<gblocks, T, 0, stream>>>(x, agg0, agg1, Bcat1, bias1, h1, N);

  // ---- layer 2
  fill_zero<<<1024, T, 0, stream>>>(agg0, 2 * nf);
  scatter_feat<<<sblocks, T, 0, stream>>>(edges,         edges + E,     h1, agg0, E);
  scatter_feat<<<sblocks, T, 0, stream>>>(edges + 2 * E, edges + 3 * E, h1, agg1, E);
  scale_rows<<<fblocks, T, 0, stream>>>(agg0, dinv,     nf);
  scale_rows<<<fblocks, T, 0, stream>>>(agg1, dinv + N, nf);
  sage_gemm<64, false><<<gblocks, T, 0, stream>>>(h1, agg0, agg1, Bcat2, bias2, h2, N);

  // ---- scoring on relation-0 edges and negative edges
  edge_score<<<sblocks, T, 0, stream>>>(edges, edges + E, h2, out, E);
  edge_score<<<sblocks, T, 0, stream>>>(negE,  negE + E,  h2, out + E, E);
}